// LinearAttention_35321811042456
// MI455X (gfx1250) — compile-verified
//
#include <hip/hip_runtime.h>
#include <math.h>

typedef __attribute__((ext_vector_type(2))) float v2f;
typedef __attribute__((ext_vector_type(8))) float v8f;

constexpr int B_ = 16, C_ = 128, W_ = 16384, H_ = 4, D_ = 32;
constexpr int NCHUNK = 16, CHUNK = W_ / NCHUNK;   // 1024 cols per chunk
constexpr int RT = 64;                            // round tile (cols in LDS)
constexpr int NROUND = CHUNK / RT;                // 16
constexpr int PART_STRIDE = 32 + 32 * 32;         // l[32] + ctx[32][32]
constexpr size_t CTXF_OFF = (size_t)B_ * H_ * NCHUNK * PART_STRIDE; // floats
constexpr int NT2 = 64;                           // kernel-2 column tile
constexpr int XP = 68;                            // K1 sX pitch: 16B-aligned rows
constexpr float SCALE_ = 0.17677669529663687f;    // 32^-0.5
constexpr float EPS_ = 1e-5f;

__device__ __forceinline__ v8f wmma4(v2f a, v2f b, v8f c) {
  // D(16x16,f32) = A(16x4,f32) * B(4x16,f32) + C  -- exact fp32 matrix op
  return __builtin_amdgcn_wmma_f32_16x16x4_f32(false, a, false, b, (short)0, c,
                                               false, false);
}

// Flat generic pointers into LDS carry the LDS byte offset in addr[31:0]
// (ISA 10.2 aperture mapping), so the low dword is the wave-relative dsaddr.
__device__ __forceinline__ unsigned lds_off(const void* p) {
  return (unsigned)(unsigned long long)p;
}

// Async global->LDS copy engine (tracked by ASYNCcnt): 16B per lane, no VGPR
// round trip. VDST = LDS byte address, VADDR = 64-bit global address (GV mode).
__device__ __forceinline__ void async_ld_b128(unsigned lds_byte, const float* g) {
  asm volatile("global_load_async_to_lds_b128 %0, %1, off"
               :: "v"(lds_byte), "v"(g) : "memory");
}
__device__ __forceinline__ void wait_async0() {
  asm volatile("s_wait_asynccnt 0x0" ::: "memory");
}

// -------------------------------------------------------------------------
// Kernel 1: per (b,h,chunk) compute partial k-softmax sums l_d and the
// unnormalized context ctx[d][e] = sum_n exp(k_dn) * v_en over this chunk.
// Double-buffered async global->LDS staging overlaps HBM fetch with WMMA.
// -------------------------------------------------------------------------
__global__ __launch_bounds__(256) void lin_attn_ctx_partial(
    const float* __restrict__ x, const float* __restrict__ Wqkv,
    float* __restrict__ ws) {
  const int bh = blockIdx.x / NCHUNK;
  const int chunk = blockIdx.x % NCHUNK;
  const int b = bh / H_, h = bh % H_;
  const int t = threadIdx.x;
  const int lane = t & 31, wv = t >> 5;
  const int l = lane & 15;
  const int hiOff = (lane >> 4) << 1;   // 0 | 2 : K-half for A/B frags
  const int rowOff = (lane >> 4) << 3;  // 0 | 8 : M-half for C/D frags

  __shared__ float sW[64 * 130];     // k|v weight rows (padded stride)
  __shared__ float sX[2][128 * XP];  // double-buffered x column tile
  __shared__ float sKV[64 * 66];     // exp(k) rows 0..31, v rows 32..63
  __shared__ float sL[32];

  // stage W_k / W_v rows for this head: rows 0..31 = K, 32..63 = V
  for (int i = t; i < 64 * 32; i += 256) {
    int r = i >> 5, c4 = (i & 31) * 4;
    int g = (r < 32) ? (C_ + h * D_ + r) : (2 * C_ + h * D_ + (r - 32));
    float4 v = *(const float4*)(Wqkv + g * C_ + c4);
    float* d = &sW[r * 130 + c4];
    d[0] = v.x; d[1] = v.y; d[2] = v.z; d[3] = v.w;
  }
  if (t < 32) sL[t] = 0.f;
  float ctxAcc[4] = {0.f, 0.f, 0.f, 0.f};

  const float* xb = x + (size_t)b * C_ * W_;
  // prologue: kick off round 0's tile into buffer 0
  {
    const int n0 = chunk * CHUNK;
    for (int i = t; i < 128 * 16; i += 256) {
      int r = i >> 4, c4 = (i & 15) * 4;
      async_ld_b128(lds_off(&sX[0][r * XP + c4]),
                    xb + (size_t)r * W_ + n0 + c4);
    }
  }

  for (int rd = 0; rd < NROUND; ++rd) {
    const int buf = rd & 1;
    wait_async0();       // my async writes into sX[buf] have landed
    __syncthreads();     // ... and everyone else's; prior round fully consumed

    if (rd + 1 < NROUND) {  // prefetch next tile into the other buffer
      const int n1 = chunk * CHUNK + (rd + 1) * RT;
      for (int i = t; i < 128 * 16; i += 256) {
        int r = i >> 4, c4 = (i & 15) * 4;
        async_ld_b128(lds_off(&sX[buf ^ 1][r * XP + c4]),
                      xb + (size_t)r * W_ + n1 + c4);
      }
    }

    // 64xRT output = (4 mTiles) x (4 nTiles); 2 tiles per wave
    for (int tt = wv * 2; tt < wv * 2 + 2; ++tt) {
      const int m = tt >> 2, nt = tt & 3;
      v8f acc;
#pragma unroll
      for (int r = 0; r < 8; ++r) acc[r] = 0.f;
      const float* aB = &sW[(m * 16 + l) * 130 + hiOff];
      const float* bB = &sX[buf][hiOff * XP + nt * 16 + l];
#pragma unroll 8
      for (int c = 0; c < 32; ++c) {  // K = 128 in steps of 4
        v2f a;  a[0] = aB[4 * c];            a[1] = aB[4 * c + 1];
        v2f bb; bb[0] = bB[(4 * c) * XP];    bb[1] = bB[(4 * c + 1) * XP];
        acc = wmma4(a, bb, acc);
      }
      const int col = nt * 16 + l;
#pragma unroll
      for (int r = 0; r < 8; ++r) {
        int row = m * 16 + rowOff + r;
        float v = acc[r];
        if (row < 32) v = __expf(v);  // softmax numerator for k rows
        sKV[row * 66 + col] = v;
      }
    }
    __syncthreads();

    if (t < 32) {  // l_d += sum_n exp(k_dn)   (fixed order, deterministic)
      float s = 0.f;
      const float* p = &sKV[t * 66];
      for (int j = 0; j < RT; ++j) s += p[j];
      sL[t] += s;
    }
    {  // 1024 ctx entries, 4 per thread, accumulate in registers
      const int base = t * 4;
#pragma unroll
      for (int i = 0; i < 4; ++i) {
        int idx = base + i, d = idx >> 5, e = idx & 31;
        const float* pk = &sKV[d * 66];
        const float* pv = &sKV[(32 + e) * 66];
        float s = 0.f;
        for (int j = 0; j < RT; ++j) s += pk[j] * pv[j];
        ctxAcc[i] += s;
      }
    }
    __syncthreads();  // sKV consumers done before next round overwrites
  }

  float* part = ws + (size_t)blockIdx.x * PART_STRIDE;
  if (t < 32) part[t] = sL[t];
#pragma unroll
  for (int i = 0; i < 4; ++i) part[32 + t * 4 + i] = ctxAcc[i];
}

// -------------------------------------------------------------------------
// Kernel 1b: fold chunk partials; normalize by (sum exp) * W  (k-softmax, v/w)
// -------------------------------------------------------------------------
__global__ __launch_bounds__(256) void lin_attn_ctx_reduce(
    float* __restrict__ ws) {
  const int bh = blockIdx.x;
  const int t = threadIdx.x;
  __shared__ float sLs[32];
  const float* base = ws + (size_t)bh * NCHUNK * PART_STRIDE;
  if (t < 32) {
    float s = 0.f;
    for (int c = 0; c < NCHUNK; ++c) s += base[(size_t)c * PART_STRIDE + t];
    sLs[t] = s;
  }
  __syncthreads();
  float* dst = ws + CTXF_OFF + (size_t)bh * 1024;
  for (int i = t; i < 1024; i += 256) {
    int d = i >> 5;
    float s = 0.f;
    for (int c = 0; c < NCHUNK; ++c)
      s += base[(size_t)c * PART_STRIDE + 32 + i];
    dst[i] = s / (sLs[d] * (float)W_);
  }
}

// -------------------------------------------------------------------------
// Kernel 2: per (b, 64-col tile): q GEMM -> d-softmax*SCALE -> ctx apply ->
// W_out GEMM + bias -> channel LayerNorm -> write. All GEMMs via fp32 WMMA.
// -------------------------------------------------------------------------
__global__ __launch_bounds__(256) void lin_attn_out(
    const float* __restrict__ x, const float* __restrict__ Wqkv,
    const float* __restrict__ Wout, const float* __restrict__ bout,
    const float* __restrict__ gamma, const float* __restrict__ beta,
    const float* __restrict__ ws, float* __restrict__ out) {
  const int b = blockIdx.x >> 8;       // 256 tiles per batch
  const int n0 = (blockIdx.x & 255) * NT2;
  const int t = threadIdx.x;
  const int lane = t & 31, wv = t >> 5;
  const int l = lane & 15;
  const int hiOff = (lane >> 4) << 1;
  const int rowOff = (lane >> 4) << 3;

  __shared__ float sA[128 * 130];      // Wq, then W_out
  __shared__ float sX[128 * 66];       // x tile, then head-mixed "out"
  __shared__ float sQ[128 * 66];       // exp(q), then y
  __shared__ float sC[4][32 * 34];     // per-head 32x32 context
  __shared__ float sBias[128], sG[128], sBt[128];
  __shared__ float sMu[NT2], sRs[NT2];

  for (int i = t; i < 128 * 32; i += 256) {  // Wq = rows 0..127 of W_qkv
    int r = i >> 5, c4 = (i & 31) * 4;
    float4 v = *(const float4*)(Wqkv + r * C_ + c4);
    float* d = &sA[r * 130 + c4];
    d[0] = v.x; d[1] = v.y; d[2] = v.z; d[3] = v.w;
  }
  const float* xb = x + (size_t)b * C_ * W_;
  for (int i = t; i < 128 * 16; i += 256) {
    int r = i >> 4, c4 = (i & 15) * 4;
    float4 v = *(const float4*)(xb + (size_t)r * W_ + n0 + c4);
    float* d = &sX[r * 66 + c4];
    d[0] = v.x; d[1] = v.y; d[2] = v.z; d[3] = v.w;
  }
  {
    const float* cf = ws + CTXF_OFF + (size_t)(b * H_) * 1024;
    for (int i = t; i < 4096; i += 256) {
      int h = i >> 10, de = i & 1023;
      sC[h][(de >> 5) * 34 + (de & 31)] = cf[i];
    }
  }
  if (t < 128) { sBias[t] = bout[t]; sG[t] = gamma[t]; sBt[t] = beta[t]; }
  __syncthreads();

  // ---- q = Wq @ x_tile; store exp(q). 8 mTiles x 4 nTiles, 4 tiles/wave.
  for (int tt = wv * 4; tt < wv * 4 + 4; ++tt) {
    const int m = tt >> 2, nt = tt & 3;
    v8f acc;
#pragma unroll
    for (int r = 0; r < 8; ++r) acc[r] = 0.f;
    const float* aB = &sA[(m * 16 + l) * 130 + hiOff];
    const float* bB = &sX[hiOff * 66 + nt * 16 + l];
#pragma unroll 8
    for (int c = 0; c < 32; ++c) {
      v2f a;  a[0] = aB[4 * c];         a[1] = aB[4 * c + 1];
      v2f bb; bb[0] = bB[(4 * c) * 66]; bb[1] = bB[(4 * c + 1) * 66];
      acc = wmma4(a, bb, acc);
    }
    const int col = nt * 16 + l;
#pragma unroll
    for (int r = 0; r < 8; ++r)
      sQ[(m * 16 + rowOff + r) * 66 + col] = __expf(acc[r]);
  }
  __syncthreads();

  // ---- softmax over head-dim (32 rows per head) per column, * SCALE
  {
    int col = t & 63, h = t >> 6;
    float s = 0.f;
    for (int d = 0; d < 32; ++d) s += sQ[(h * 32 + d) * 66 + col];
    float sc = SCALE_ / s;
    for (int d = 0; d < 32; ++d) sQ[(h * 32 + d) * 66 + col] *= sc;
  }
  __syncthreads();

  // ---- out[h*32+e, n] = sum_d ctx[h][d][e] * q[h*32+d, n]  (K=32 WMMA)
  for (int tt = wv * 4; tt < wv * 4 + 4; ++tt) {
    const int m = tt >> 2, nt = tt & 3;
    const int h = m >> 1, e0 = (m & 1) * 16;
    v8f acc;
#pragma unroll
    for (int r = 0; r < 8; ++r) acc[r] = 0.f;
    const float* aB = &sC[h][hiOff * 34 + e0 + l];  // A[M=e][K=d] = ctx[d][e]
    const float* bB = &sQ[(h * 32 + hiOff) * 66 + nt * 16 + l];
#pragma unroll
    for (int c = 0; c < 8; ++c) {
      v2f a;  a[0] = aB[(4 * c) * 34];  a[1] = aB[(4 * c + 1) * 34];
      v2f bb; bb[0] = bB[(4 * c) * 66]; bb[1] = bB[(4 * c + 1) * 66];
      acc = wmma4(a, bb, acc);
    }
    const int col = nt * 16 + l;
#pragma unroll
    for (int r = 0; r < 8; ++r)
      sX[(m * 16 + rowOff + r) * 66 + col] = acc[r];  // reuse sX as "out"
  }
  __syncthreads();

  for (int i = t; i < 128 * 32; i += 256) {  // swap weights: W_out
    int r = i >> 5, c4 = (i & 31) * 4;
    float4 v = *(const float4*)(Wout + r * C_ + c4);
    float* d = &sA[r * 130 + c4];
    d[0] = v.x; d[1] = v.y; d[2] = v.z; d[3] = v.w;
  }
  __syncthreads();

  // ---- y = W_out @ out + bias  -> sQ
  for (int tt = wv * 4; tt < wv * 4 + 4; ++tt) {
    const int m = tt >> 2, nt = tt & 3;
    v8f acc;
#pragma unroll
    for (int r = 0; r < 8; ++r) acc[r] = 0.f;
    const float* aB = &sA[(m * 16 + l) * 130 + hiOff];
    const float* bB = &sX[hiOff * 66 + nt * 16 + l];
#pragma unroll 8
    for (int c = 0; c < 32; ++c) {
      v2f a;  a[0] = aB[4 * c];         a[1] = aB[4 * c + 1];
      v2f bb; bb[0] = bB[(4 * c) * 66]; bb[1] = bB[(4 * c + 1) * 66];
      acc = wmma4(a, bb, acc);
    }
    const int col = nt * 16 + l;
#pragma unroll
    for (int r = 0; r < 8; ++r) {
      int row = m * 16 + rowOff + r;
      sQ[row * 66 + col] = acc[r] + sBias[row];
    }
  }
  __syncthreads();

  // ---- channel LayerNorm (biased var over 128), then gamma/beta, write
  if (t < NT2) {
    float mu = 0.f;
    for (int ch = 0; ch < 128; ++ch) mu += sQ[ch * 66 + t];
    mu *= (1.f / 128.f);
    float var = 0.f;
    for (int ch = 0; ch < 128; ++ch) {
      float d = sQ[ch * 66 + t] - mu;
      var += d * d;
    }
    sMu[t] = mu;
    sRs[t] = rsqrtf(var * (1.f / 128.f) + EPS_);
  }
  __syncthreads();
  for (int i = t; i < 128 * 16; i += 256) {
    int ch = i >> 4, cb = (i & 15) * 4;
    float g = sG[ch], bt = sBt[ch];
    float4 o;
    o.x = (sQ[ch * 66 + cb + 0] - sMu[cb + 0]) * sRs[cb + 0] * g + bt;
    o.y = (sQ[ch * 66 + cb + 1] - sMu[cb + 1]) * sRs[cb + 1] * g + bt;
    o.z = (sQ[ch * 66 + cb + 2] - sMu[cb + 2]) * sRs[cb + 2] * g + bt;
    o.w = (sQ[ch * 66 + cb + 3] - sMu[cb + 3]) * sRs[cb + 3] * g + bt;
    *(float4*)(out + (size_t)(b * C_ + ch) * W_ + n0 + cb) = o;
  }
}

extern "C" void kernel_launch(void* const* d_in, const int* in_sizes, int n_in,
                              void* d_out, int out_size, void* d_ws,
                              size_t ws_size, hipStream_t stream) {
  (void)in_sizes; (void)n_in; (void)out_size; (void)ws_size;
  const float* x     = (const float*)d_in[0];
  const float* Wqkv  = (const float*)d_in[1];
  const float* Wout  = (const float*)d_in[2];
  const float* bout  = (const float*)d_in[3];
  const float* gamma = (const float*)d_in[4];
  const float* beta  = (const float*)d_in[5];
  float* ws  = (float*)d_ws;   // uses CTXF_OFF + 64*1024 floats (~4.6 MB)
  float* out = (float*)d_out;

  lin_attn_ctx_partial<<<B_ * H_ * NCHUNK, 256, 0, stream>>>(x, Wqkv, ws);
  lin_attn_ctx_reduce<<<B_ * H_, 256, 0, stream>>>(ws);
  lin_attn_out<<<B_ * (W_ / NT2), 256, 0, stream>>>(x, Wqkv, Wout, bout,
                                                    gamma, beta, ws, out);
}